// QNLPModel_83683142795294
// MI455X (gfx1250) — compile-verified
//
#include <hip/hip_runtime.h>

#define H 384
#define S 96
#define HF (2 * H) // float columns of a complex HxH matrix (re,im interleaved)

typedef float v2f __attribute__((ext_vector_type(2)));
typedef float v8f __attribute__((ext_vector_type(8)));

// workspace layout (float offsets)
#define WS_OUTL 0                    // 768 x 16 : V_l^T @ [e0, 1, 0...]
#define WS_OUTR (WS_OUTL + HF * 16)  // 768 x 16 : V_r^T @ [ReDw, ReDb, ImDw, ImDb, 0...]
#define WS_OUTUL (WS_OUTR + HF * 16) // 384 x 16 : u_l = U_l @ t_l  (cols: re0,re1,im0,im1)
#define WS_OUTUR (WS_OUTUL + H * 16) // 384 x 16 : u_r = U_r @ t_r
#define WS_G (WS_OUTUR + H * 16)     // 96 x 4   : {p.re,p.im,q.re,q.im} per token
#define WS_SUMS (WS_G + S * 4)       // 4        : {Sw.re,Sw.im,Sb.re,Sb.im}

__device__ __forceinline__ float warp_sum(float v) {
  for (int o = 16; o > 0; o >>= 1) v += __shfl_xor(v, o, 32);
  return v;
}

// ---------------------------------------------------------------------------
// Phase 1:
//  blocks 0..5  : OutL = Vl_f^T @ R_l   (WMMA f32 16x16x4, R_l staged in LDS)
//  blocks 6..11 : OutR = Vr_f^T @ R_r
//  blocks 12..23: per-token dots p_t = Dw.emb_t, q_t = Db.emb_t
//  block  24    : Sw = sum(Dw), Sb = sum(Db)
// ---------------------------------------------------------------------------
__global__ __launch_bounds__(256) void k_phase1(
    const int* __restrict__ x, const float* __restrict__ E,
    const float* __restrict__ Dw, const float* __restrict__ Db,
    const float* __restrict__ Vl, const float* __restrict__ Vr,
    float* __restrict__ ws) {
  __shared__ float Rs[H * 16]; // 24 KB: block-local RHS (384 x 16)
  const int gwave = blockIdx.x * 8 + (threadIdx.x >> 5);
  const int lane = threadIdx.x & 31;

  if (blockIdx.x < 12) {
    const bool isL = blockIdx.x < 6; // block-uniform role
    // ---- one-time cooperative RHS build (all divergence lives here) ----
    const long e0off = (long)x[0] * H;
    for (int idx = threadIdx.x; idx < H * 16; idx += 256) {
      const int k = idx >> 4, n = idx & 15;
      float v = 0.0f;
      if (isL) { // columns [emb0, ones, 0...]
        if (n == 0) v = E[e0off + k];
        else if (n == 1) v = 1.0f;
      } else {   // columns [Re Dw, Re Db, Im Dw, Im Db, 0...]
        if (n == 0) v = Dw[2 * k];
        else if (n == 1) v = Db[2 * k];
        else if (n == 2) v = Dw[2 * k + 1];
        else if (n == 3) v = Db[2 * k + 1];
      }
      Rs[idx] = v;
    }
    __syncthreads();

    const int m0 = (isL ? gwave : gwave - 48) * 16;
    const float* __restrict__ Vf = isL ? Vl : Vr;
    float* __restrict__ Out = ws + (isL ? WS_OUTL : WS_OUTR);

    const int ml = lane & 15;       // M (or N) index within tile
    const int kb = (lane >> 4) * 2; // K sub-base {0,2}
    const float* ap = Vf + kb * HF + m0 + ml; // A[m,k] = Vf[k, m0+m]
    const float* bp = Rs + kb * 16 + ml;

    v8f acc = {};
    v2f a, b;
    a.x = ap[0]; a.y = ap[HF];
    b.x = bp[0]; b.y = bp[16];
    for (int kk = 0; kk < H - 4; kk += 4) { // software-pipelined, branch-free
      v2f a2, b2;
      a2.x = ap[4 * HF]; a2.y = ap[5 * HF];
      b2.x = bp[64];     b2.y = bp[80];
      acc = __builtin_amdgcn_wmma_f32_16x16x4_f32(false, a, false, b,
                                                  (short)0, acc, false, false);
      a = a2; b = b2;
      ap += 4 * HF; bp += 64;
    }
    acc = __builtin_amdgcn_wmma_f32_16x16x4_f32(false, a, false, b,
                                                (short)0, acc, false, false);
    // D layout: VGPR r -> row r (lanes 0-15) / row r+8 (lanes 16-31), col = ml
    const int rbase = (lane >> 4) * 8;
    for (int r = 0; r < 8; ++r)
      Out[(m0 + rbase + r) * 16 + ml] = acc[r];
  } else if (blockIdx.x < 24) {
    const int t = gwave - 96;
    const float* __restrict__ e = E + (long)x[t] * H;
    float pr = 0, pi = 0, qr = 0, qi = 0;
    for (int j = lane; j < H; j += 32) {
      const float ej = e[j];
      pr += ej * Dw[2 * j];  pi += ej * Dw[2 * j + 1];
      qr += ej * Db[2 * j];  qi += ej * Db[2 * j + 1];
    }
    pr = warp_sum(pr); pi = warp_sum(pi); qr = warp_sum(qr); qi = warp_sum(qi);
    if (lane == 0) {
      float* g = ws + WS_G + t * 4;
      g[0] = pr; g[1] = pi; g[2] = qr; g[3] = qi;
    }
  } else if ((threadIdx.x >> 5) == 0) { // block 24, wave 0: sums of Dw, Db
    float swr = 0, swi = 0, sbr = 0, sbi = 0;
    for (int j = lane; j < H; j += 32) {
      swr += Dw[2 * j]; swi += Dw[2 * j + 1];
      sbr += Db[2 * j]; sbi += Db[2 * j + 1];
    }
    swr = warp_sum(swr); swi = warp_sum(swi);
    sbr = warp_sum(sbr); sbi = warp_sum(sbi);
    if (lane == 0) {
      float* sm = ws + WS_SUMS;
      sm[0] = swr; sm[1] = swi; sm[2] = sbr; sm[3] = sbi;
    }
  }
}

// Generated Z column for phase 2 (complex arithmetic folded into real GEMM).
// conjMode=false: t = relu(s) .* (V^T X)          (left density path)
// conjMode=true : t = relu(s) .* (V^T conj(Y))    (right density path)
// Z columns: n in {0,1} -> Re(u) cols, {2,3} -> Im(u) cols; else 0.
__device__ __forceinline__ float zval(const float* __restrict__ Out,
                                      const float* __restrict__ sv,
                                      bool conjMode, int ii, int n) {
  if (n >= 4) return 0.0f;
  const int i = ii >> 1, par = ii & 1, c = n & 1;
  float r = sv[i];
  r = r > 0.0f ? r : 0.0f; // relu
  float tre, tim;
  if (!conjMode) {
    tre = r * Out[(2 * i) * 16 + c];
    tim = r * Out[(2 * i + 1) * 16 + c];
  } else {
    tre = r * (Out[(2 * i) * 16 + c] + Out[(2 * i + 1) * 16 + 2 + c]);
    tim = r * (Out[(2 * i + 1) * 16 + c] - Out[(2 * i) * 16 + 2 + c]);
  }
  if (n < 2) return par ? -tim : tre; // Re-result column
  return par ? tre : tim;             // Im-result column
}

// ---------------------------------------------------------------------------
// Phase 2: OutU[k,n] = sum_ii Uf[k,ii] * Z(ii,n)   (M=384, K=768, WMMA f32)
//  blocks 0..2 : U_l path -> OutUL ; blocks 3..5 : U_r path -> OutUR
//  Z (768 x 16) staged once in LDS per block -> branch-free inner loop.
// ---------------------------------------------------------------------------
__global__ __launch_bounds__(256) void k_phase2(
    const float* __restrict__ Ul, const float* __restrict__ Ur,
    const float* __restrict__ sl, const float* __restrict__ sr,
    float* __restrict__ ws) {
  __shared__ float Zs[HF * 16]; // 48 KB
  const bool isL = blockIdx.x < 3; // block-uniform role
  const float* __restrict__ Out = ws + (isL ? WS_OUTL : WS_OUTR);
  const float* __restrict__ sv = isL ? sl : sr;

  for (int idx = threadIdx.x; idx < HF * 16; idx += 256) {
    Zs[idx] = zval(Out, sv, !isL, idx >> 4, idx & 15);
  }
  __syncthreads();

  const int gwave = blockIdx.x * 8 + (threadIdx.x >> 5);
  const int m0 = (isL ? gwave : gwave - 24) * 16;
  const float* __restrict__ Uf = isL ? Ul : Ur;
  float* __restrict__ OutU = ws + (isL ? WS_OUTUL : WS_OUTUR);

  const int lane = threadIdx.x & 31;
  const int ml = lane & 15;
  const int kb = (lane >> 4) * 2;
  // A row is contiguous in K -> single b64 load per step
  const v2f* __restrict__ ap = (const v2f*)(Uf + (long)(m0 + ml) * HF + kb);
  const float* bp = Zs + kb * 16 + ml;

  v8f acc = {};
  v2f a, b;
  a = ap[0];
  b.x = bp[0]; b.y = bp[16];
  for (int kk = 0; kk < HF - 4; kk += 4) { // software-pipelined
    v2f a2, b2;
    a2 = ap[2];
    b2.x = bp[64]; b2.y = bp[80];
    acc = __builtin_amdgcn_wmma_f32_16x16x4_f32(false, a, false, b,
                                                (short)0, acc, false, false);
    a = a2; b = b2;
    ap += 2; bp += 64;
  }
  acc = __builtin_amdgcn_wmma_f32_16x16x4_f32(false, a, false, b,
                                              (short)0, acc, false, false);
  const int rbase = (lane >> 4) * 8;
  for (int r = 0; r < 8; ++r)
    OutU[(m0 + rbase + r) * 16 + ml] = acc[r];
}

struct cpx { float re, im; };
__device__ __forceinline__ cpx cmul(cpx a, cpx b) {
  return {a.re * b.re - a.im * b.im, a.re * b.im + a.im * b.re};
}
__device__ __forceinline__ cpx cadd(cpx a, cpx b) { return {a.re + b.re, a.im + b.im}; }

// ---------------------------------------------------------------------------
// Phase 3: A2 = X^T u_l, B2 = Y^T u_r (deterministic LDS reductions),
//          W = G_1 * ... * G_{S-1} (2x2 chain), phi = tr(A2 W B2 W^H)
// ---------------------------------------------------------------------------
__global__ __launch_bounds__(256) void k_phase3(
    const int* __restrict__ x, const float* __restrict__ E,
    const float* __restrict__ Dw, const float* __restrict__ Db,
    const float* __restrict__ ws, float* __restrict__ out, int out_size) {
  __shared__ float red[16][257];
  __shared__ float fin[16];
  const float* __restrict__ OutUL = ws + WS_OUTUL;
  const float* __restrict__ OutUR = ws + WS_OUTUR;
  const int tid = threadIdx.x;

  float part[16];
#pragma unroll
  for (int s = 0; s < 16; ++s) part[s] = 0.0f;

  const float* __restrict__ e0 = E + (long)x[0] * H;
  for (int k = tid; k < H; k += 256) {
    const float xk[2] = {e0[k], 1.0f};
#pragma unroll
    for (int r = 0; r < 2; ++r)
#pragma unroll
      for (int c = 0; c < 2; ++c) {
        part[r * 4 + c * 2 + 0] += xk[r] * OutUL[k * 16 + c];     // Re A2
        part[r * 4 + c * 2 + 1] += xk[r] * OutUL[k * 16 + 2 + c]; // Im A2
      }
    const float yre[2] = {Dw[2 * k], Db[2 * k]};
    const float yim[2] = {Dw[2 * k + 1], Db[2 * k + 1]};
#pragma unroll
    for (int r = 0; r < 2; ++r)
#pragma unroll
      for (int c = 0; c < 2; ++c) {
        const float ure = OutUR[k * 16 + c], uim = OutUR[k * 16 + 2 + c];
        part[8 + r * 4 + c * 2 + 0] += yre[r] * ure - yim[r] * uim;
        part[8 + r * 4 + c * 2 + 1] += yre[r] * uim + yim[r] * ure;
      }
  }
#pragma unroll
  for (int s = 0; s < 16; ++s) red[s][tid] = part[s];
  __syncthreads();
  if (tid < 16) { // fixed-order serial sum: bit-deterministic
    float a = 0.0f;
    for (int j = 0; j < 256; ++j) a += red[tid][j];
    fin[tid] = a;
  }
  __syncthreads();

  if (tid == 0) {
    cpx A2[2][2], B2[2][2];
    for (int r = 0; r < 2; ++r)
      for (int c = 0; c < 2; ++c) {
        A2[r][c] = {fin[r * 4 + c * 2 + 0], fin[r * 4 + c * 2 + 1]};
        B2[r][c] = {fin[8 + r * 4 + c * 2 + 0], fin[8 + r * 4 + c * 2 + 1]};
      }
    const float* sm = ws + WS_SUMS;
    const cpx Sw = {sm[0], sm[1]}, Sb = {sm[2], sm[3]};
    cpx W00 = {1, 0}, W01 = {0, 0}, W10 = {0, 0}, W11 = {1, 0};
    for (int t = S - 1; t >= 1; --t) { // W = G_1 (G_2 (... G_{S-1}))
      const float* g = ws + WS_G + t * 4;
      const cpx p = {g[0], g[1]}, q = {g[2], g[3]};
      const cpx n00 = cadd(cmul(p, W00), cmul(Sw, W10));
      const cpx n01 = cadd(cmul(p, W01), cmul(Sw, W11));
      const cpx n10 = cadd(cmul(q, W00), cmul(Sb, W10));
      const cpx n11 = cadd(cmul(q, W01), cmul(Sb, W11));
      W00 = n00; W01 = n01; W10 = n10; W11 = n11;
    }
    // T1 = A2 * W ; T2 = B2 * W^H ; phi = tr(T1 * T2)
    const cpx Wm[2][2] = {{W00, W01}, {W10, W11}};
    cpx Wh[2][2];
    for (int r = 0; r < 2; ++r)
      for (int c = 0; c < 2; ++c) Wh[r][c] = {Wm[c][r].re, -Wm[c][r].im};
    cpx T1[2][2], T2[2][2];
    for (int r = 0; r < 2; ++r)
      for (int c = 0; c < 2; ++c) {
        T1[r][c] = cadd(cmul(A2[r][0], Wm[0][c]), cmul(A2[r][1], Wm[1][c]));
        T2[r][c] = cadd(cmul(B2[r][0], Wh[0][c]), cmul(B2[r][1], Wh[1][c]));
      }
    cpx phi = {0, 0};
    for (int r = 0; r < 2; ++r)
      for (int c = 0; c < 2; ++c) phi = cadd(phi, cmul(T1[r][c], T2[c][r]));
    out[0] = phi.re;
    if (out_size > 1) out[1] = phi.im;
  }
}

extern "C" void kernel_launch(void* const* d_in, const int* in_sizes, int n_in,
                              void* d_out, int out_size, void* d_ws, size_t ws_size,
                              hipStream_t stream) {
  const int* x = (const int*)d_in[0];
  const float* E = (const float*)d_in[1];
  const float* Dw = (const float*)d_in[2]; // complex64 -> interleaved float
  const float* Db = (const float*)d_in[3];
  const float* Ul = (const float*)d_in[4];
  const float* sl = (const float*)d_in[5];
  const float* Vl = (const float*)d_in[6];
  const float* Ur = (const float*)d_in[7];
  const float* sr = (const float*)d_in[8];
  const float* Vr = (const float*)d_in[9];
  float* ws = (float*)d_ws;
  float* out = (float*)d_out;

  k_phase1<<<25, 256, 0, stream>>>(x, E, Dw, Db, Vl, Vr, ws);
  k_phase2<<<6, 256, 0, stream>>>(Ul, Ur, sl, sr, ws);
  k_phase3<<<1, 256, 0, stream>>>(x, E, Dw, Db, ws, out, out_size);
}